// AvgPooling_similarity_3453153706427
// MI455X (gfx1250) — compile-verified
//
#include <hip/hip_runtime.h>

typedef __attribute__((ext_vector_type(2))) float v2f;
typedef __attribute__((ext_vector_type(8))) float v8f;

// ---------------------------------------------------------------------------
// Kernel 1: x_out[j,:] = 0.5*(x[2j,:] + x[2j+1,:]) via V_WMMA_F32_16X16X4_F32.
//
// D(16x16) = P(16x32) * X(32x16), P[m,k] = 0.5 if k in {2m, 2m+1}.
// K=32 is done as 8 chained K=4 WMMAs. For slice i (K columns 4i..4i+3):
//   A_i nonzeros: A_i[2i][0..1] = A_i[2i+1][2..3] = 0.5
//   A layout (16x4 f32): lanes 0-15 = M, VGPR0 = {K0 | K2}, VGPR1 = {K1 | K3}
//   -> both A VGPRs hold 0.5 at lanes (2i) and (16 + 2i + 1), zero elsewhere.
//   B layout (4x16 f32) mirrored: VGPR0 = rows {K0 | K2}, VGPR1 = rows {K1 | K3},
//   N = lane&15.
// One block = 8 waves = 32 input rows x 128 cols; wave w owns cols 16w..16w+15.
// ---------------------------------------------------------------------------
__global__ __launch_bounds__(256) void x_pair_mean_wmma(const float* __restrict__ x,
                                                        float* __restrict__ xo) {
  const int tid  = threadIdx.x;
  const int lane = tid & 31;
  const int wave = tid >> 5;

  const int r0 = blockIdx.x * 32;   // first input row of this block
  const int p0 = blockIdx.x * 16;   // first output (pair) row
  const int c0 = wave * 16;         // column stripe for this wave
  const int lc = lane & 15;
  const bool hi = (lane >= 16);

  v8f acc = {};
#pragma unroll
  for (int i = 0; i < 8; ++i) {
    // Constant A slice: 0.5 at lanes 2i and 2i+17 (both VGPRs identical).
    const float av = (lane == 2 * i || lane == 2 * i + 17) ? 0.5f : 0.0f;
    v2f a;
    a.x = av;
    a.y = av;

    // B slice: rows r0+4i+{0,1} (lanes 0-15) / r0+4i+{2,3} (lanes 16-31).
    const int rbase = r0 + 4 * i + (hi ? 2 : 0);
    v2f b;
    b.x = x[(size_t)(rbase + 0) * 128 + c0 + lc];
    b.y = x[(size_t)(rbase + 1) * 128 + c0 + lc];

    acc = __builtin_amdgcn_wmma_f32_16x16x4_f32(
        /*neg_a=*/false, a, /*neg_b=*/false, b,
        /*c_mod=*/(short)0, acc, /*reuse_a=*/false, /*reuse_b=*/false);
  }

  // C/D layout: VGPR j -> row p0+j (lanes 0-15) / p0+8+j (lanes 16-31).
  const int prow = p0 + (hi ? 8 : 0);
#pragma unroll
  for (int j = 0; j < 8; ++j) {
    xo[(size_t)(prow + j) * 128 + c0 + lc] = acc[j];
  }
}

// ---------------------------------------------------------------------------
// Kernel 2: pos / seq / ori / batch paths (tiny vs. x; plain vectorized VALU).
// One thread per output pair j. Rows of 3 floats: two consecutive rows are
// 6 contiguous floats starting at 6j -> three aligned float2 loads per array.
// ---------------------------------------------------------------------------
__global__ __launch_bounds__(256) void small_pair_kernel(
    const float* __restrict__ pos, const int* __restrict__ seq,
    const float* __restrict__ ori, const int* __restrict__ batch,
    float* __restrict__ pos_o, float* __restrict__ seq_o,
    float* __restrict__ ori_o, float* __restrict__ batch_o, int npairs) {
  const int j = blockIdx.x * blockDim.x + threadIdx.x;
  if (j >= npairs) return;

  // pos mean
  const float2* pos2 = (const float2*)pos;
  float2 q0 = pos2[3 * (size_t)j + 0];
  float2 q1 = pos2[3 * (size_t)j + 1];
  float2 q2 = pos2[3 * (size_t)j + 2];
  pos_o[3 * (size_t)j + 0] = 0.5f * (q0.x + q1.y);
  pos_o[3 * (size_t)j + 1] = 0.5f * (q0.y + q2.x);
  pos_o[3 * (size_t)j + 2] = 0.5f * (q1.x + q2.y);

  // ori mean + L2 normalize
  const float2* ori2 = (const float2*)ori;
  float2 o0 = ori2[3 * (size_t)j + 0];
  float2 o1 = ori2[3 * (size_t)j + 1];
  float2 o2 = ori2[3 * (size_t)j + 2];
  float mx = 0.5f * (o0.x + o1.y);
  float my = 0.5f * (o0.y + o2.x);
  float mz = 0.5f * (o1.x + o2.y);
  float nrm = sqrtf(mx * mx + my * my + mz * mz);
  float inv = 1.0f / fmaxf(nrm, 1e-12f);
  ori_o[3 * (size_t)j + 0] = mx * inv;
  ori_o[3 * (size_t)j + 1] = my * inv;
  ori_o[3 * (size_t)j + 2] = mz * inv;

  // seq: max(seq[2j]//2, seq[2j+1]//2)  (>>1 == floor-div)
  int s0 = seq[2 * (size_t)j] >> 1;
  int s1 = seq[2 * (size_t)j + 1] >> 1;
  seq_o[j] = (float)(s0 > s1 ? s0 : s1);

  // batch: segment max
  int b0 = batch[2 * (size_t)j];
  int b1 = batch[2 * (size_t)j + 1];
  batch_o[j] = (float)(b0 > b1 ? b0 : b1);
}

extern "C" void kernel_launch(void* const* d_in, const int* in_sizes, int n_in,
                              void* d_out, int out_size, void* d_ws, size_t ws_size,
                              hipStream_t stream) {
  (void)n_in; (void)out_size; (void)d_ws; (void)ws_size;

  const float* x     = (const float*)d_in[0];
  const float* pos   = (const float*)d_in[1];
  const int*   seq   = (const int*)d_in[2];
  const float* ori   = (const float*)d_in[3];
  const int*   batch = (const int*)d_in[4];

  const int n      = in_sizes[0] / 128;  // N rows
  const int npairs = n / 2;

  float* out     = (float*)d_out;
  float* xo      = out;
  float* pos_o   = xo + (size_t)npairs * 128;
  float* seq_o   = pos_o + (size_t)npairs * 3;
  float* ori_o   = seq_o + (size_t)npairs;
  float* batch_o = ori_o + (size_t)npairs * 3;

  // x path: one block per 32 input rows (N is a multiple of 32).
  x_pair_mean_wmma<<<n / 32, 256, 0, stream>>>(x, xo);

  // small arrays: one thread per pair.
  small_pair_kernel<<<(npairs + 255) / 256, 256, 0, stream>>>(
      pos, seq, ori, batch, pos_o, seq_o, ori_o, batch_o, npairs);
}